// Net_20899310862685
// MI455X (gfx1250) — compile-verified
//
#include <hip/hip_runtime.h>
#include <stdint.h>

// ---------------------------------------------------------------------------
// 2-layer GraphConv (DGL norm='both') for MI455X / gfx1250.
//  - SpMM via edge-parallel f32 global atomics (L2-resident: 25.6MB features
//    << 192MB L2; ~1.2GB total L2 atomic traffic dominates runtime).
//  - Dense [N,64]@[64,64] GEMM via V_WMMA_F32_16X16X4_F32 (exact f32 math).
//    W is staged into LDS *transposed* once per workgroup by the Tensor Data
//    Mover (3-D descriptor: X=1 elem, Y=k walks a W column via
//    tensor_dim0_stride=64, Z=n via tensor_dim1_stride=1), with TDM padding
//    (4 dwords per 64 -> row stride 68) so each ds_load_b64 of a B pair hits
//    64 distinct LDS banks: bank = (4*idx + k + t) mod 64, a permutation.
//  - Each wave computes a 16-row x 64-col output strip: A fragment loaded
//    once per k-step, fed to 4 WMMAs (4 accumulators); B pairs are
//    contiguous in LDS (Wt[n][ka..ka+1]) -> single ds_load_b64, no shuffles.
// ---------------------------------------------------------------------------

typedef __attribute__((ext_vector_type(2))) float v2f;
typedef __attribute__((ext_vector_type(8))) float v8f;
typedef __attribute__((ext_vector_type(4))) unsigned int v4u;
typedef __attribute__((ext_vector_type(4))) int v4i;
typedef __attribute__((ext_vector_type(8))) int v8i;

#define GCN_N 100000
#define GCN_E 1200000
#define GCN_D 64
#define LDS_STRIDE 68   // 64 dwords + 4 dwords TDM pad -> conflict-free b64 reads

__global__ void fill_zero_kernel(float* __restrict__ p, int n) {
    int i = blockIdx.x * blockDim.x + threadIdx.x;
    if (i < n) p[i] = 0.0f;
}

__global__ void degree_kernel(const int* __restrict__ src,
                              const int* __restrict__ dst,
                              float* __restrict__ deg_out,
                              float* __restrict__ deg_in, int e) {
    int i = blockIdx.x * blockDim.x + threadIdx.x;
    if (i < e) {
        atomicAdd(&deg_out[src[i]], 1.0f);
        atomicAdd(&deg_in[dst[i]], 1.0f);
    }
}

// deg -> rsqrt(max(deg,1)) in place for both arrays
__global__ void norm_kernel(float* __restrict__ deg_out,
                            float* __restrict__ deg_in, int n) {
    int i = blockIdx.x * blockDim.x + threadIdx.x;
    if (i < n) {
        deg_out[i] = rsqrtf(fmaxf(deg_out[i], 1.0f));
        deg_in[i]  = rsqrtf(fmaxf(deg_in[i], 1.0f));
    }
}

// agg[dst] += x[src] * norm_out[src]; 16 threads per edge, float4 per thread.
// Index/norm loads are same-address within contiguous lane groups -> the
// memory system coalesces them to single requests; atomics are the cost.
__global__ void scatter_kernel(const float* __restrict__ x,
                               const float* __restrict__ norm_out,
                               const int* __restrict__ src,
                               const int* __restrict__ dst,
                               float* __restrict__ agg, int e) {
    int t = blockIdx.x * blockDim.x + threadIdx.x;
    int eidx = t >> 4;
    if (eidx >= e) return;
    int c = (t & 15) << 2;
    int s = src[eidx];
    int d = dst[eidx];
    float ns = norm_out[s];
    const float4 v = *(const float4*)(x + s * GCN_D + c);
    float* ap = agg + d * GCN_D + c;
    atomicAdd(ap + 0, v.x * ns);
    atomicAdd(ap + 1, v.y * ns);
    atomicAdd(ap + 2, v.z * ns);
    atomicAdd(ap + 3, v.w * ns);
}

// out[m,:] = act( (agg[m,:] * nin[m]) @ W + bias )
// One wave32 per 16-row strip, all 64 output cols (4 x 16x16 WMMA tiles).
// A layout (16x4 f32, 2 VGPRs): lanes 0-15 -> K={k,k+1} of row m0+lane,
// lanes 16-31 -> K={k+2,k+3}. B (4x16): same K half-split, N across lanes.
// C/D (8 VGPRs): m = m0 + r + 8*(lane>=16), n = n0 + (lane&15).
__global__ __launch_bounds__(256) void gemm_wmma_kernel(
    const float* __restrict__ agg, const float* __restrict__ nin,
    const float* __restrict__ W, const float* __restrict__ bias,
    float* __restrict__ out, int nrows, int do_relu) {
    __shared__ float Wt[GCN_D * LDS_STRIDE];   // W^T, 17 KB, TDM-padded rows

    const int wave = threadIdx.x >> 5;
    const int lane = threadIdx.x & 31;

    if (wave == 0) {
        // Tensor Data Mover: DMA W (64x64 f32 row-major in global) into LDS
        // transposed. 3-D tile: X=1 element, Y=k (global stride 64 elems),
        // Z=n (global stride 1 elem). LDS written linearly with a 4-dword pad
        // every 64 dwords -> Wt[n][k] at word n*68 + k.
        const uint64_t ga = (uint64_t)(const void*)W;
        const unsigned lds_off = (unsigned)(uintptr_t)&Wt[0]; // LDS byte offset
        // D# group0: count=1 | lds_addr | global_addr[56:0] | type=2.
        v4u g0;
        g0[0] = 1u;
        g0[1] = lds_off;
        g0[2] = (unsigned)(ga & 0xFFFFFFFFu);
        g0[3] = (unsigned)((ga >> 32) & 0x01FFFFFFu) | 0x80000000u; // type=2
        // D# group1: data_size=2 (4B) | pad_enable | pad_interval=5 (64 dw)
        //            | pad_amount=3 (4 dw) | dims/strides/tile dims.
        v8i g1;
        g1[0] = 0x00020000 | (1 << 20) | (5 << 22) | (3 << 25);
        g1[1] = (GCN_D << 16);          // tensor_dim0 = 64 (low 16)
        g1[2] = (GCN_D << 16);          // tensor_dim0 hi=0 | tensor_dim1 = 64
        g1[3] = (1 << 16);              // tensor_dim1 hi=0 | tile_dim0 = 1
        g1[4] = GCN_D | (GCN_D << 16);  // tile_dim1 = 64 | tile_dim2 = 64
        g1[5] = GCN_D;                  // tensor_dim0_stride = 64 (k step)
        g1[6] = (1 << 16);              // dim0_stride hi=0 | dim1_stride = 1 (n step)
        g1[7] = 0;                      // dim1_stride hi = 0
        // D# group2: tensor_dim2 = 64; dim3/tile_dim3/dim2_stride = 0.
        v4i g2 = {GCN_D, 0, 0, 0};
        v4i g3 = {0, 0, 0, 0};
#if __clang_major__ >= 23
        v8i z8 = {0, 0, 0, 0, 0, 0, 0, 0};
        __builtin_amdgcn_tensor_load_to_lds(g0, g1, g2, g3, z8, 0);
#else
        __builtin_amdgcn_tensor_load_to_lds(g0, g1, g2, g3, 0);
#endif
        __builtin_amdgcn_s_wait_tensorcnt(0);
    }
    __syncthreads();   // all waves reach this before any tail-guard return

    const int mt = blockIdx.x * 8 + wave;
    if (mt * 16 >= nrows) return;

    const int half = lane >> 4;       // 0: K pair {k,k+1}, 1: {k+2,k+3}
    const int idx  = lane & 15;
    const int m0   = mt * 16;
    const int row  = m0 + idx;

    const float ni = nin[row];        // fold D_in^{-1/2} into A
    const float* __restrict__ arow = agg + row * GCN_D;

    v8f acc[4] = {v8f{}, v8f{}, v8f{}, v8f{}};
#pragma unroll
    for (int k = 0; k < GCN_D; k += 4) {
        const int ka = k + half * 2;
        v2f a;
        a.x = arow[ka + 0] * ni;
        a.y = arow[ka + 1] * ni;
#pragma unroll
        for (int nt = 0; nt < 4; ++nt) {
            // Contiguous pair Wt[n][ka..ka+1] -> single ds_load_b64,
            // 8-byte aligned (ka, LDS_STRIDE both even), bank-conflict-free.
            const v2f b = *(const v2f*)&Wt[(nt * 16 + idx) * LDS_STRIDE + ka];
            acc[nt] = __builtin_amdgcn_wmma_f32_16x16x4_f32(
                /*neg_a=*/false, a, /*neg_b=*/false, b,
                /*c_mod=*/(short)0, acc[nt],
                /*reuse_a=*/false, /*reuse_b=*/false);
        }
    }

#pragma unroll
    for (int nt = 0; nt < 4; ++nt) {
        const float bv = bias[nt * 16 + idx];
#pragma unroll
        for (int r = 0; r < 8; ++r) {
            const int m = m0 + r + half * 8;
            float v = acc[nt][r] + bv;
            if (do_relu) v = fmaxf(v, 0.0f);
            out[m * GCN_D + nt * 16 + idx] = v;
        }
    }
}

extern "C" void kernel_launch(void* const* d_in, const int* in_sizes, int n_in,
                              void* d_out, int out_size, void* d_ws, size_t ws_size,
                              hipStream_t stream) {
    const float* feats = (const float*)d_in[0];   // [N, 64]
    const int*   src   = (const int*)d_in[1];     // [E]
    const int*   dst   = (const int*)d_in[2];     // [E]
    const float* W1    = (const float*)d_in[3];   // [64, 64]
    const float* b1    = (const float*)d_in[4];   // [64]
    const float* W2    = (const float*)d_in[5];   // [64, 64]
    const float* b2    = (const float*)d_in[6];   // [64]
    float* out = (float*)d_out;                   // [N, 64]

    // Workspace layout (floats): norm_out[N] | norm_in[N] | agg[N*64] | x1[N*64]
    float* ws   = (float*)d_ws;
    float* nout = ws;
    float* nin  = ws + GCN_N;
    float* agg  = ws + 2 * GCN_N;
    float* x1   = agg + (size_t)GCN_N * GCN_D;

    const int nagg = GCN_N * GCN_D;
    const int scat_threads = GCN_E * 16;
    const int mtiles = GCN_N / 16;                // 6250 (exact)
    const int gemm_blocks = (mtiles + 7) / 8;     // 8 waves (strips) per block

    // Degrees + norms (recomputed every call: deterministic).
    fill_zero_kernel<<<(2 * GCN_N + 255) / 256, 256, 0, stream>>>(nout, 2 * GCN_N);
    degree_kernel<<<(GCN_E + 255) / 256, 256, 0, stream>>>(src, dst, nout, nin, GCN_E);
    norm_kernel<<<(GCN_N + 255) / 256, 256, 0, stream>>>(nout, nin, GCN_N);

    // Layer 1: agg = A_hat * feats ; x1 = relu(agg @ W1 + b1)
    fill_zero_kernel<<<(nagg + 255) / 256, 256, 0, stream>>>(agg, nagg);
    scatter_kernel<<<(scat_threads + 255) / 256, 256, 0, stream>>>(
        feats, nout, src, dst, agg, GCN_E);
    gemm_wmma_kernel<<<gemm_blocks, 256, 0, stream>>>(
        agg, nin, W1, b1, x1, GCN_N, 1);

    // Layer 2: agg = A_hat * x1 ; out = agg @ W2 + b2
    fill_zero_kernel<<<(nagg + 255) / 256, 256, 0, stream>>>(agg, nagg);
    scatter_kernel<<<(scat_threads + 255) / 256, 256, 0, stream>>>(
        x1, nout, src, dst, agg, GCN_E);
    gemm_wmma_kernel<<<gemm_blocks, 256, 0, stream>>>(
        agg, nin, W2, b2, out, GCN_N, 0);
}